// ReprojectionMultiRigModelFixedRel_68839735820966
// MI455X (gfx1250) — compile-verified
//
#include <hip/hip_runtime.h>
#include <stdint.h>

// Unaligned-friendly vector types (rows are only 4-byte aligned: 28B / 12B strides).
typedef float v2f  __attribute__((ext_vector_type(2)));
typedef int   v2i  __attribute__((ext_vector_type(2)));
typedef float v2fu __attribute__((ext_vector_type(2), aligned(4)));
typedef float v4fu __attribute__((ext_vector_type(4), aligned(4)));

struct F3 { float x, y, z; };

__device__ __forceinline__ F3 cross3(F3 a, F3 b) {
    F3 r;
    r.x = a.y * b.z - a.z * b.y;
    r.y = a.z * b.x - a.x * b.z;
    r.z = a.x * b.y - a.y * b.x;
    return r;
}

// v + w*(2*cross(qv,v)) + cross(qv, 2*cross(qv,v))
__device__ __forceinline__ F3 quat_rotate(float qx, float qy, float qz, float qw, F3 v) {
    F3 qv; qv.x = qx; qv.y = qy; qv.z = qz;
    F3 t = cross3(qv, v);
    t.x *= 2.0f; t.y *= 2.0f; t.z *= 2.0f;
    F3 c = cross3(qv, t);
    F3 r;
    r.x = v.x + qw * t.x + c.x;
    r.y = v.y + qw * t.y + c.y;
    r.z = v.z + qw * t.z + c.z;
    return r;
}

// LDS layout: [0..55] rel_poses (8x7), [56..63] intrs (8), [64..79] camera_pps (8x2)
#define TAB_REL   0
#define TAB_INTR  56
#define TAB_PP    64
#define TAB_SIZE  80

__global__ __launch_bounds__(256) void reproj_multirig_kernel(
    const float* __restrict__ points_2d,   // (N,2)
    const int*   __restrict__ cam_idx,     // (N,)
    const int*   __restrict__ grp_idx,     // (N,2) [group, member]
    const int*   __restrict__ pt_idx,      // (N,)
    const float* __restrict__ camera_pps,  // (8,2)
    const float* __restrict__ rel_poses,   // (8,7)
    const float* __restrict__ intrs,       // (8,1)
    const float* __restrict__ points_3d,   // (NUM_PTS,3)
    const float* __restrict__ ref_poses,   // (NUM_GROUPS,7)
    float*       __restrict__ out,         // (N,2)
    int n)
{
    __shared__ float tab[TAB_SIZE];

    // --- Stage tiny tables into LDS with CDNA5 async global->LDS DMA ---
    {
        int t = threadIdx.x;
        if (t < TAB_SIZE) {
            const float* src;
            if (t < TAB_INTR)       src = rel_poses  + t;
            else if (t < TAB_PP)    src = intrs      + (t - TAB_INTR);
            else                    src = camera_pps + (t - TAB_PP);
            uint32_t lds_addr = (uint32_t)(uintptr_t)(&tab[t]);
            asm volatile("global_load_async_to_lds_b32 %0, %1, off"
                         :: "v"(lds_addr), "v"(src)
                         : "memory");
        }
        asm volatile("s_wait_asynccnt 0" ::: "memory");
    }
    __syncthreads();

    int i = blockIdx.x * blockDim.x + threadIdx.x;
    if (i >= n) return;

    // --- Streamed inputs: non-temporal (read-once; keep L2 for the gather tables) ---
    v2i gi  = __builtin_nontemporal_load((const v2i*)grp_idx + i);
    int cam = __builtin_nontemporal_load(cam_idx + i);
    int pt  = __builtin_nontemporal_load(pt_idx + i);
    v2f p2  = __builtin_nontemporal_load((const v2f*)points_2d + i);

    int g = gi.x;
    int m = gi.y;

    // --- ref pose gather (112 KB table, L2/WGP$-resident): 28B row, 4B aligned ---
    const float* rp = ref_poses + 7 * g;
    v4fu r0 = *(const v4fu*)rp;         // t.x t.y t.z q.x
    v2fu r1 = *(const v2fu*)(rp + 4);   // q.y q.z
    float rqw = rp[6];                  // q.w
    float rtx = r0.x, rty = r0.y, rtz = r0.z;
    float rqx = r0.w, rqy = r1.x, rqz = r1.y;

    // --- rel pose from LDS (ds_load) ---
    const float* lp = &tab[TAB_REL + 7 * m];
    float atx = lp[0], aty = lp[1], atz = lp[2];
    float aqx = lp[3], aqy = lp[4], aqz = lp[5], aqw = lp[6];

    // pose_t = rel.t + rotate(rel.q, ref.t)
    F3 reft; reft.x = rtx; reft.y = rty; reft.z = rtz;
    F3 rt = quat_rotate(aqx, aqy, aqz, aqw, reft);
    float ptx = atx + rt.x;
    float pty = aty + rt.y;
    float ptz = atz + rt.z;

    // pose_q = quat_mul(rel.q, ref.q)   (x,y,z,w layout)
    float qx = aqw * rqx + aqx * rqw + aqy * rqz - aqz * rqy;
    float qy = aqw * rqy - aqx * rqz + aqy * rqw + aqz * rqx;
    float qz = aqw * rqz + aqx * rqy - aqy * rqx + aqz * rqw;
    float qw = aqw * rqw - aqx * rqx - aqy * rqy - aqz * rqz;

    // --- 3D point gather (6 MB table, L2-resident): 12B row, 4B aligned ---
    const float* pr = points_3d + 3 * pt;
    v2fu pxy = *(const v2fu*)pr;
    float pz = pr[2];
    F3 P; P.x = pxy.x; P.y = pxy.y; P.z = pz;

    F3 pc = quat_rotate(qx, qy, qz, qw, P);
    pc.x += ptx; pc.y += pty; pc.z += ptz;

    // --- intrinsics from LDS ---
    float f  = tab[TAB_INTR + cam];
    float cx = tab[TAB_PP + 2 * cam];
    float cy = tab[TAB_PP + 2 * cam + 1];

    float invz = 1.0f / pc.z;   // single IEEE divide, shared by u and v
    float u = f * pc.x * invz + cx - p2.x;
    float v = f * pc.y * invz + cy - p2.y;

    v2f o; o.x = u; o.y = v;
    __builtin_nontemporal_store(o, (v2f*)out + i);
}

extern "C" void kernel_launch(void* const* d_in, const int* in_sizes, int n_in,
                              void* d_out, int out_size, void* d_ws, size_t ws_size,
                              hipStream_t stream) {
    const float* points_2d  = (const float*)d_in[0];
    const int*   cam_idx    = (const int*)  d_in[1];
    const int*   grp_idx    = (const int*)  d_in[2];
    const int*   pt_idx     = (const int*)  d_in[3];
    const float* camera_pps = (const float*)d_in[4];
    const float* rel_poses  = (const float*)d_in[5];
    const float* intrs      = (const float*)d_in[6];
    const float* points_3d  = (const float*)d_in[7];
    const float* ref_poses  = (const float*)d_in[8];
    float*       out        = (float*)d_out;

    int n = in_sizes[1];  // camera_indices length == N observations
    const int block = 256;
    int grid = (n + block - 1) / block;

    reproj_multirig_kernel<<<grid, block, 0, stream>>>(
        points_2d, cam_idx, grp_idx, pt_idx,
        camera_pps, rel_poses, intrs, points_3d, ref_poses,
        out, n);
}